// LiteNTK_45561013076149
// MI455X (gfx1250) — compile-verified
//
#include <hip/hip_runtime.h>
#include <hip/hip_bf16.h>
#include <math.h>

// LiteNTK on MI455X (gfx1250): algebraically-collapsed NTK.
//  - diag recursion needs only row norms (diagonal S clips to 0.9999 const)
//  - sandwich einsums collapse to bilinear forms p_N^T ntk p_M / p_N^T ntk r_M
//  - pairwise sigma = X_N X_M^T via V_WMMA_F32_16X16X4_F32, fused with the
//    arccos recursion and bilinear reduction (no 85MB intermediates).
//  - division in the recursion replaced by hardware v_rcp_f32 (tmp > 0,
//    result is clamped to +-0.9999 before acos -> 1-ulp rcp is harmless).

typedef __attribute__((ext_vector_type(2))) float v2f;
typedef __attribute__((ext_vector_type(8))) float v8f;

#define NG   96     // graphs (source == target count)
#define NN   48     // nodes per graph
#define RK   16     // low-rank
#define DD   128    // feature dim
#define NCLS 10     // classes
#define PI_F 3.14159265358979f
#define IPI_F 0.318309886183791f
#define LDSW 132    // padded LDS row stride (floats), 132%64=4 -> conflict-free

// ---------------- per-source-graph precompute: diag + p vector -------------
__global__ __launch_bounds__(64) void ntk_prep_src(
    const float* __restrict__ U, const float* __restrict__ V,
    const float* __restrict__ X, float* __restrict__ dS, float* __restrict__ pS)
{
    const int g = blockIdx.x, tid = threadIdx.x;
    __shared__ float u[RK], w[RK];
    __shared__ float VU[RK][RK], VUp[RK][RK];

    // layer-1 diag scale: S on diagonal clips to 0.9999 (constant)
    const float s0 = 0.9999f;
    const float th0 = acosf(s0);
    const float c0  = (s0 * (PI_F - th0) + sqrtf(1.f - s0 * s0)) * IPI_F;
    const float sc1 = sqrtf(c0);

    // row norms -> diag layers 0 and 1
    for (int a = tid; a < NN; a += 64) {
        const float* xr = X + (size_t)g * NN * DD + (size_t)a * DD;
        float s = 0.f;
        #pragma unroll 4
        for (int d = 0; d < DD; ++d) s += xr[d] * xr[d];
        float d0 = sqrtf(s + 1e-4f);
        dS[g * (2 * NN) + 0 * NN + a] = d0;
        dS[g * (2 * NN) + 1 * NN + a] = d0 * sc1;
    }
    // u = column sums of U (rank-vector)
    if (tid < RK) {
        float s = 0.f;
        for (int a = 0; a < NN; ++a) s += U[(size_t)g * NN * RK + a * RK + tid];
        u[tid] = s;
    }
    // VU = V @ U (16x16)
    for (int e = tid; e < RK * RK; e += 64) {
        int i = e >> 4, j = e & 15;
        float s = 0.f;
        for (int k = 0; k < NN; ++k)
            s += V[(size_t)g * RK * NN + i * NN + k] * U[(size_t)g * NN * RK + k * RK + j];
        VU[i][j] = s;
    }
    __syncthreads();
    // VUp = VU @ VU
    for (int e = tid; e < RK * RK; e += 64) {
        int i = e >> 4, j = e & 15;
        float s = 0.f;
        for (int k = 0; k < RK; ++k) s += VU[i][k] * VU[k][j];
        VUp[i][j] = s;
    }
    __syncthreads();
    // w = VUp^T u
    if (tid < RK) {
        float s = 0.f;
        for (int i = 0; i < RK; ++i) s += VUp[i][tid] * u[i];
        w[tid] = s;
    }
    __syncthreads();
    // p = V^T w (48-vector)
    for (int a = tid; a < NN; a += 64) {
        float s = 0.f;
        for (int b = 0; b < RK; ++b) s += V[(size_t)g * RK * NN + b * NN + a] * w[b];
        pS[g * NN + a] = s;
    }
}

// ---------------- per-target-graph precompute: diag + r vector -------------
__global__ __launch_bounds__(64) void ntk_prep_tgt(
    const float* __restrict__ A, const float* __restrict__ X,
    float* __restrict__ dT, float* __restrict__ rT)
{
    const int g = blockIdx.x, tid = threadIdx.x;
    __shared__ float r0[NN];

    const float s0 = 0.9999f;
    const float th0 = acosf(s0);
    const float c0  = (s0 * (PI_F - th0) + sqrtf(1.f - s0 * s0)) * IPI_F;
    const float sc1 = sqrtf(c0);

    for (int a = tid; a < NN; a += 64) {
        const float* xr = X + (size_t)g * NN * DD + (size_t)a * DD;
        float s = 0.f;
        #pragma unroll 4
        for (int d = 0; d < DD; ++d) s += xr[d] * xr[d];
        float d0 = sqrtf(s + 1e-4f);
        dT[g * (2 * NN) + 0 * NN + a] = d0;
        dT[g * (2 * NN) + 1 * NN + a] = d0 * sc1;
    }
    // r0 = (A + 1e-4 I)^T 1  (column sums)
    if (tid < NN) {
        float s = 0.f;
        for (int d = 0; d < NN; ++d) s += A[(size_t)g * NN * NN + d * NN + tid];
        r0[tid] = s + 1e-4f;
    }
    __syncthreads();
    // r = (A + 1e-4 I)^T r0
    if (tid < NN) {
        float s = 0.f;
        for (int d = 0; d < NN; ++d) s += A[(size_t)g * NN * NN + d * NN + tid] * r0[d];
        rT[g * NN + tid] = s + 1e-4f * r0[tid];
    }
}

// ---------------- fused pair kernel: sigma (WMMA f32) + recursion + bilinear
// one workgroup per (N,M) pair; 9 waves, one 16x16 sigma tile each.
__global__ __launch_bounds__(288) void ntk_pair(
    const float* __restrict__ XL, const float* __restrict__ XR,
    const float* __restrict__ dL, const float* __restrict__ dR,
    const float* __restrict__ pL, const float* __restrict__ pR,
    float* __restrict__ out)
{
    const int M = blockIdx.x;   // right graph
    const int N = blockIdx.y;   // left graph
    const int tid = threadIdx.x;

    __shared__ __align__(16) float lXa[NN * LDSW];
    __shared__ __align__(16) float lXb[NN * LDSW];
    __shared__ float ldL[2 * NN], ldR[2 * NN];
    __shared__ float lp[NN], lr[NN];
    __shared__ float red[9];

    // stage X tiles (48x128 each) into LDS, float4, padded stride
    const float4* xa4 = (const float4*)(XL + (size_t)N * NN * DD);
    const float4* xb4 = (const float4*)(XR + (size_t)M * NN * DD);
    for (int i = tid; i < NN * DD / 4; i += 288) {
        int row = i >> 5;            // 32 float4 per row
        int c4  = (i & 31) << 2;
        *(float4*)&lXa[row * LDSW + c4] = xa4[i];
        *(float4*)&lXb[row * LDSW + c4] = xb4[i];
    }
    if (tid < 2 * NN) { ldL[tid] = dL[N * 2 * NN + tid]; ldR[tid] = dR[M * 2 * NN + tid]; }
    else if (tid < 2 * NN + NN)       lp[tid - 2 * NN]      = pL[N * NN + tid - 2 * NN];
    else if (tid < 2 * NN + 2 * NN)   lr[tid - 3 * NN]      = pR[M * NN + tid - 3 * NN];
    __syncthreads();

    const int w    = tid >> 5;
    const int lane = tid & 31;
    const int tr   = w / 3, tc = w % 3;
    const int g    = lane >> 4;       // half-wave group
    const int ln   = lane & 15;
    const float* Ar = &lXa[(tr * 16 + ln) * LDSW];
    const float* Br = &lXb[(tc * 16 + ln) * LDSW];

    // sigma tile = X_N[tile_r] @ X_M[tile_c]^T over K=128: 32 x WMMA f32 16x16x4
    v8f acc = {0.f, 0.f, 0.f, 0.f, 0.f, 0.f, 0.f, 0.f};
    #pragma unroll 4
    for (int k = 0; k < DD / 4; ++k) {
        const int kb = k * 4 + g * 2;
        v2f a; a.x = Ar[kb]; a.y = Ar[kb + 1];
        v2f b; b.x = Br[kb]; b.y = Br[kb + 1];
        acc = __builtin_amdgcn_wmma_f32_16x16x4_f32(
            false, a, false, b, (short)0, acc, false, false);
    }

    // arccos-NTK recursion + bilinear reduction, per accumulator element
    const int nc = tc * 16 + ln;
    const float prc = lr[nc];
    float part = 0.f;
    #pragma unroll
    for (int v = 0; v < 8; ++v) {
        const int m = tr * 16 + v + g * 8;
        float sig = acc[v] + 1e-4f;
        float ntk = sig;
        #pragma unroll
        for (int l = 0; l < 2; ++l) {
            float tmp = ldL[l * NN + m] * ldR[l * NN + nc] + 1e-6f;
            // tmp > 0 and S is clamped before acos -> raw v_rcp_f32 is fine
            float S = sig * __builtin_amdgcn_rcpf(tmp);
            S = fminf(fmaxf(S, -0.9999f), 0.9999f);
            float th  = acosf(S);
            float pmt = PI_F - th;
            float dsv = pmt * IPI_F;
            float sn  = (S * pmt + sqrtf(fmaxf(1.f - S * S, 0.f))) * IPI_F;
            ntk = ntk * dsv + sn;
            sig = sn * tmp;
        }
        part += lp[m] * ntk * prc;
    }
    // wave32 reduce, then 9-wave block reduce
    #pragma unroll
    for (int off = 16; off > 0; off >>= 1) part += __shfl_down(part, off);
    if (lane == 0) red[w] = part;
    __syncthreads();
    if (tid == 0) {
        float s = 0.f;
        #pragma unroll
        for (int i = 0; i < 9; ++i) s += red[i];
        out[N * NG + M] = s;
    }
}

// ---------------- tiny solve: alpha = (K_SS + reg I)^{-1} y; pred = K_ST^T a
__global__ __launch_bounds__(128) void ntk_solve(
    const float* __restrict__ Kss, const float* __restrict__ Kst,
    const float* __restrict__ yS, float* __restrict__ pred)
{
    const int tid = threadIdx.x;
    __shared__ float Kl[NG * (NG + 1)];
    __shared__ float ay[NG * NCLS];

    for (int i = tid; i < NG * NG; i += 128) {
        int r = i / NG, c = i % NG;
        Kl[r * (NG + 1) + c] = Kss[i];
    }
    for (int i = tid; i < NG * NCLS; i += 128) ay[i] = yS[i];
    __syncthreads();
    if (tid == 0) {
        float t = 0.f;
        for (int i = 0; i < NG; ++i) t += Kl[i * (NG + 1) + i];
        float reg = 1e-6f * t / (float)NG;
        for (int i = 0; i < NG; ++i) Kl[i * (NG + 1) + i] += reg;
    }
    // Gaussian elimination (SPD + reg -> no pivoting)
    for (int k = 0; k < NG; ++k) {
        __syncthreads();
        const float inv = 1.0f / Kl[k * (NG + 1) + k];
        for (int i = k + 1 + tid; i < NG; i += 128) {
            float f = Kl[i * (NG + 1) + k] * inv;
            for (int j = k; j < NG; ++j)
                Kl[i * (NG + 1) + j] -= f * Kl[k * (NG + 1) + j];
            #pragma unroll
            for (int c = 0; c < NCLS; ++c)
                ay[i * NCLS + c] -= f * ay[k * NCLS + c];
        }
    }
    // back substitution (alpha kept in ay)
    for (int k = NG - 1; k >= 0; --k) {
        __syncthreads();
        if (tid < NCLS) {
            float s = ay[k * NCLS + tid];
            for (int j = k + 1; j < NG; ++j)
                s -= Kl[k * (NG + 1) + j] * ay[j * NCLS + tid];
            ay[k * NCLS + tid] = s / Kl[k * (NG + 1) + k];
        }
    }
    __syncthreads();
    // pred[t,c] = sum_s K_ST[s,t] * alpha[s,c]
    for (int i = tid; i < NG * NCLS; i += 128) {
        int t = i / NCLS, c = i % NCLS;
        float s = 0.f;
        for (int sg = 0; sg < NG; ++sg) s += Kst[sg * NG + t] * ay[sg * NCLS + c];
        pred[i] = s;
    }
}

extern "C" void kernel_launch(void* const* d_in, const int* in_sizes, int n_in,
                              void* d_out, int out_size, void* d_ws, size_t ws_size,
                              hipStream_t stream) {
    const float* U_S = (const float*)d_in[0];   // (96,48,16)
    const float* V_S = (const float*)d_in[1];   // (96,16,48)
    const float* X_S = (const float*)d_in[2];   // (96,48,128)
    const float* y_S = (const float*)d_in[3];   // (96,10)
    const float* A_T = (const float*)d_in[4];   // (96,48,48)
    const float* X_T = (const float*)d_in[5];   // (96,48,128)

    float* out  = (float*)d_out;                // [pred 960 | K_SS 9216]
    float* pred = out;
    float* K_SS = out + NG * NCLS;

    float* ws  = (float*)d_ws;
    float* dS  = ws;                            // 96*2*48
    float* pS  = dS + NG * 2 * NN;              // 96*48
    float* dT  = pS + NG * NN;                  // 96*2*48
    float* rT  = dT + NG * 2 * NN;              // 96*48
    float* Kst = rT + NG * NN;                  // 96*96

    ntk_prep_src<<<NG, 64, 0, stream>>>(U_S, V_S, X_S, dS, pS);
    ntk_prep_tgt<<<NG, 64, 0, stream>>>(A_T, X_T, dT, rT);

    dim3 grid(NG, NG);
    ntk_pair<<<grid, 288, 0, stream>>>(X_S, X_S, dS, dS, pS, pS, K_SS);
    ntk_pair<<<grid, 288, 0, stream>>>(X_S, X_T, dS, dT, pS, rT, Kst);

    ntk_solve<<<1, 128, 0, stream>>>(K_SS, Kst, y_S, pred);
}